// HADGNN_14388140441985
// MI455X (gfx1250) — compile-verified
//
#include <hip/hip_runtime.h>
#include <math.h>

typedef __attribute__((ext_vector_type(2))) float v2f;
typedef __attribute__((ext_vector_type(8))) float v8f;

#define T_      8
#define N_      5000
#define E_      160000
#define IN_DIM_ 64
#define HID_    128
#define H_      8
#define D_      16
#define ETOT_   (E_ + N_)
#define EPS_    1e-5f

// ---------------------------------------------------------------------------
// Generic fp32 WMMA GEMM:  C[M x Ncols] = act(A[M x K] @ B + bias)
//   TRANSB = false : B is K x Ncols, element (k, col) at B[k*ldb + col]
//   TRANSB = true  : logical B(k,col) stored at B[col*ldb + k]   (A @ W^T)
// Block = 128 threads = 4 waves; each wave owns a 16x16 tile of C.
// grid.x = ceil(M/16), grid.y = Ncols/64.
// Out-of-range rows are CLAMPED (not predicated): clamped rows only feed C
// rows that are never stored, so the k-loop is branch-free and keeps EXEC
// all-ones around every V_WMMA_F32_16X16X4_F32.
// ---------------------------------------------------------------------------
template <bool TRANSB, bool RELU>
__global__ __launch_bounds__(128)
void gemm_f32_wmma(const float* __restrict__ A, int lda,
                   const float* __restrict__ B, int ldb,
                   const float* __restrict__ bias,
                   float* __restrict__ C, int ldc,
                   int M, int K)
{
    const int wave = threadIdx.x >> 5;
    const int lane = threadIdx.x & 31;
    const int half = lane >> 4;       // 0: K={0,1}/rows 0..7, 1: K={2,3}/rows 8..15
    const int l16  = lane & 15;
    const int rowTile = blockIdx.x * 16;
    const int col     = blockIdx.y * 64 + wave * 16 + l16;
    const int arow    = min(rowTile + l16, M - 1);   // clamp: branch-free loads

    const float* __restrict__ Arow = A + (size_t)arow * lda + 2 * half;
    v8f acc = {};
#pragma unroll 4
    for (int k = 0; k < K; k += 4) {
        const int ka = k + 2 * half;
        const v2f a = *(const v2f*)(Arow + k);       // contiguous, 8B aligned
        v2f b;
        if (TRANSB) {
            b = *(const v2f*)(B + (size_t)col * ldb + ka);
        } else {
            b.x = B[(size_t)(ka    ) * ldb + col];
            b.y = B[(size_t)(ka + 1) * ldb + col];
        }
        // (neg_a, A, neg_b, B, c_mod, C, reuse_a, reuse_b)
        acc = __builtin_amdgcn_wmma_f32_16x16x4_f32(
                  false, a, false, b, (short)0, acc, false, false);
    }

    const float bv = bias ? bias[col] : 0.0f;
#pragma unroll
    for (int i = 0; i < 8; ++i) {
        const int r = rowTile + i + 8 * half;   // VGPR i -> row i / i+8 per lane half
        if (r < M) {
            float v = acc[i] + bv;
            if (RELU) v = fmaxf(v, 0.0f);
            C[(size_t)r * ldc + col] = v;
        }
    }
}

// ---------------------------------------------------------------------------
// Wave reductions / LayerNorm helpers (wave32)
// ---------------------------------------------------------------------------
__device__ inline float wave_sum32(float v) {
#pragma unroll
    for (int off = 16; off > 0; off >>= 1) v += __shfl_xor(v, off, 32);
    return v;
}

// y = LN(x)*g + b, one wave per row of width HID_ (lane handles 4 strided cols)
__global__ void ln_kernel(const float* __restrict__ X,
                          const float* __restrict__ g, const float* __restrict__ b,
                          float* __restrict__ Y, int rows)
{
    const int wave = threadIdx.x >> 5, lane = threadIdx.x & 31;
    const int row  = blockIdx.x * (blockDim.x >> 5) + wave;
    if (row >= rows) return;
    const float* x = X + (size_t)row * HID_;
    float v[4], s = 0.0f;
#pragma unroll
    for (int j = 0; j < 4; ++j) { v[j] = x[lane + 32 * j]; s += v[j]; }
    const float mu = wave_sum32(s) * (1.0f / HID_);
    float q = 0.0f;
#pragma unroll
    for (int j = 0; j < 4; ++j) { const float d = v[j] - mu; q += d * d; }
    const float rs = rsqrtf(wave_sum32(q) * (1.0f / HID_) + EPS_);
    float* y = Y + (size_t)row * HID_;
#pragma unroll
    for (int j = 0; j < 4; ++j) {
        const int c = lane + 32 * j;
        y[c] = (v[j] - mu) * rs * g[c] + b[c];
    }
}

// fused = LN(nf[:,T-1,:] + ta[:,T-1,:]); emb = fused + sin(n/N*6.28)*0.1 + noise*0.05
__global__ void fuse_emb_kernel(const float* __restrict__ nf, const float* __restrict__ tp,
                                const float* __restrict__ g, const float* __restrict__ b,
                                const float* __restrict__ noise, float* __restrict__ emb)
{
    const int wave = threadIdx.x >> 5, lane = threadIdx.x & 31;
    const int n = blockIdx.x * (blockDim.x >> 5) + wave;
    if (n >= N_) return;
    const size_t r = ((size_t)n * T_ + (T_ - 1)) * HID_;
    float v[4], s = 0.0f;
#pragma unroll
    for (int j = 0; j < 4; ++j) {
        const int c = lane + 32 * j;
        v[j] = nf[r + c] + tp[r + c];
        s += v[j];
    }
    const float mu = wave_sum32(s) * (1.0f / HID_);
    float q = 0.0f;
#pragma unroll
    for (int j = 0; j < 4; ++j) { const float d = v[j] - mu; q += d * d; }
    const float rs  = rsqrtf(wave_sum32(q) * (1.0f / HID_) + EPS_);
    const float div = sinf((float)n / (float)N_ * 6.28f) * 0.1f;
#pragma unroll
    for (int j = 0; j < 4; ++j) {
        const int c = lane + 32 * j;
        emb[(size_t)n * HID_ + c] =
            (v[j] - mu) * rs * g[c] + b[c] + div + noise[(size_t)n * HID_ + c] * 0.05f;
    }
}

// y = gelu_exact(LN(x)*g + b)
__global__ void ln_gelu_kernel(const float* __restrict__ X,
                               const float* __restrict__ g, const float* __restrict__ b,
                               float* __restrict__ Y, int rows)
{
    const int wave = threadIdx.x >> 5, lane = threadIdx.x & 31;
    const int row  = blockIdx.x * (blockDim.x >> 5) + wave;
    if (row >= rows) return;
    const float* x = X + (size_t)row * HID_;
    float v[4], s = 0.0f;
#pragma unroll
    for (int j = 0; j < 4; ++j) { v[j] = x[lane + 32 * j]; s += v[j]; }
    const float mu = wave_sum32(s) * (1.0f / HID_);
    float q = 0.0f;
#pragma unroll
    for (int j = 0; j < 4; ++j) { const float d = v[j] - mu; q += d * d; }
    const float rs = rsqrtf(wave_sum32(q) * (1.0f / HID_) + EPS_);
    float* y = Y + (size_t)row * HID_;
#pragma unroll
    for (int j = 0; j < 4; ++j) {
        const int c = lane + 32 * j;
        const float t = (v[j] - mu) * rs * g[c] + b[c];
        y[c] = 0.5f * t * (1.0f + erff(t * 0.70710678118654752f));
    }
}

// ---------------------------------------------------------------------------
// GAT stage kernels
// ---------------------------------------------------------------------------
__global__ void feats_init_kernel(float* __restrict__ feats,
                                  const float* __restrict__ b_gat, int n)
{
    const int tid = blockIdx.x * blockDim.x + threadIdx.x;
    if (tid < n) feats[tid] = b_gat[tid & (HID_ - 1)];
}

__global__ void attcoef_kernel(const float* __restrict__ xh,
                               const float* __restrict__ a_src, const float* __restrict__ a_dst,
                               float* __restrict__ as, float* __restrict__ ad)
{
    const int tid = blockIdx.x * blockDim.x + threadIdx.x;
    if (tid >= N_ * H_) return;
    const int h = tid % H_, n = tid / H_;
    float s1 = 0.0f, s2 = 0.0f;
#pragma unroll
    for (int d = 0; d < D_; ++d) {
        const float v = xh[(size_t)n * HID_ + h * D_ + d];
        s1 += v * a_src[h * D_ + d];
        s2 += v * a_dst[h * D_ + d];
    }
    as[tid] = s1; ad[tid] = s2;
}

__global__ void zero_ms_kernel(unsigned* __restrict__ mkey, float* __restrict__ ssum)
{
    const int tid = blockIdx.x * blockDim.x + threadIdx.x;
    if (tid < N_ * H_) { mkey[tid] = 0u; ssum[tid] = 0.0f; }
}

// order-preserving float<->uint for atomicMax of (possibly negative) floats
__device__ inline unsigned f2ord(float f) {
    const unsigned u = __float_as_uint(f);
    return (u & 0x80000000u) ? ~u : (u | 0x80000000u);
}
__device__ inline float ord2f(unsigned k) {
    const unsigned u = (k & 0x80000000u) ? (k & 0x7FFFFFFFu) : ~k;
    return __uint_as_float(u);
}

__device__ inline void edge_src_dst(const int* __restrict__ ei_t, int e, int& src, int& dst) {
    if (e < E_) { src = ei_t[e]; dst = ei_t[E_ + e]; }
    else        { src = dst = e - E_; }              // appended self-loops
}

__global__ void edge_max_kernel(const int* __restrict__ ei_t,
                                const float* __restrict__ as, const float* __restrict__ ad,
                                float* __restrict__ ebuf, unsigned* __restrict__ mkey)
{
    const int tid = blockIdx.x * blockDim.x + threadIdx.x;
    if (tid >= ETOT_ * H_) return;
    const int h = tid % H_, e = tid / H_;
    int src, dst; edge_src_dst(ei_t, e, src, dst);
    float v = as[src * H_ + h] + ad[dst * H_ + h];
    v = (v > 0.0f) ? v : 0.2f * v;                   // leaky_relu(0.2)
    ebuf[tid] = v;
    atomicMax(&mkey[dst * H_ + h], f2ord(v));
}

__global__ void edge_sum_kernel(const int* __restrict__ ei_t,
                                float* __restrict__ ebuf,
                                const unsigned* __restrict__ mkey, float* __restrict__ ssum)
{
    const int tid = blockIdx.x * blockDim.x + threadIdx.x;
    if (tid >= ETOT_ * H_) return;
    const int h = tid % H_, e = tid / H_;
    int src, dst; edge_src_dst(ei_t, e, src, dst);
    const float m  = ord2f(mkey[dst * H_ + h]);
    const float ex = expf(ebuf[tid] - m);
    ebuf[tid] = ex;
    atomicAdd(&ssum[dst * H_ + h], ex);
}

__global__ void edge_agg_kernel(const int* __restrict__ ei_t,
                                const float* __restrict__ ebuf, const float* __restrict__ ssum,
                                const float* __restrict__ xh, float* __restrict__ feats, int t)
{
    const int tid = blockIdx.x * blockDim.x + threadIdx.x;
    if (tid >= ETOT_ * H_) return;
    const int h = tid % H_, e = tid / H_;
    int src, dst; edge_src_dst(ei_t, e, src, dst);
    const float alpha = ebuf[tid] / (ssum[dst * H_ + h] + 1e-16f);
    const float* xs = xh + (size_t)src * HID_ + h * D_;
    float* fo = feats + ((size_t)dst * T_ + t) * HID_ + h * D_;
#pragma unroll
    for (int d = 0; d < D_; ++d) atomicAdd(&fo[d], xs[d] * alpha);
}

// ---------------------------------------------------------------------------
// Temporal attention: one thread per (n, h, t_q); T=8 keys, D=16
// qkv row layout: [(n*T + t)*384 + {0,128,256} + h*16 + d]
// ---------------------------------------------------------------------------
__global__ void attention_kernel(const float* __restrict__ qkv, float* __restrict__ ta)
{
    const int tid = blockIdx.x * blockDim.x + threadIdx.x;
    if (tid >= N_ * H_ * T_) return;
    const int tq = tid % T_;
    const int nh = tid / T_;
    const int h  = nh % H_;
    const int n  = nh / H_;
    const float* base = qkv + (size_t)n * T_ * 384;

    float q[D_];
#pragma unroll
    for (int d = 0; d < D_; ++d) q[d] = base[tq * 384 + h * D_ + d];

    float sc[T_];
    float mx = -1e30f;
#pragma unroll
    for (int tk = 0; tk < T_; ++tk) {
        float s = 0.0f;
#pragma unroll
        for (int d = 0; d < D_; ++d) s += q[d] * base[tk * 384 + 128 + h * D_ + d];
        s *= 0.25f;                                  // 1/sqrt(D), D=16
        sc[tk] = s;
        mx = fmaxf(mx, s);
    }
    float sum = 0.0f;
#pragma unroll
    for (int tk = 0; tk < T_; ++tk) { sc[tk] = expf(sc[tk] - mx); sum += sc[tk]; }
    const float inv = 1.0f / sum;
#pragma unroll
    for (int d = 0; d < D_; ++d) {
        float o = 0.0f;
#pragma unroll
        for (int tk = 0; tk < T_; ++tk) o += sc[tk] * base[tk * 384 + 256 + h * D_ + d];
        ta[((size_t)n * T_ + tq) * HID_ + h * D_ + d] = o * inv;
    }
}

// ---------------------------------------------------------------------------
// Final logits: NC=2, per-node dot products + sinusoidal bias on class 1
// ---------------------------------------------------------------------------
__global__ void logits_kernel(const float* __restrict__ h1g,
                              const float* __restrict__ Wc2, const float* __restrict__ bc2,
                              float* __restrict__ out)
{
    const int n = blockIdx.x * blockDim.x + threadIdx.x;
    if (n >= N_) return;
    float a0 = bc2[0], a1 = bc2[1];
    const float* x = h1g + (size_t)n * HID_;
#pragma unroll 4
    for (int k = 0; k < HID_; ++k) {
        const float xv = x[k];
        a0 += xv * Wc2[k * 2];
        a1 += xv * Wc2[k * 2 + 1];
    }
    a1 += sinf((float)n * 0.5f) * 0.2f;
    out[(size_t)n * 2]     = a0;
    out[(size_t)n * 2 + 1] = a1;
}

// ---------------------------------------------------------------------------
extern "C" void kernel_launch(void* const* d_in, const int* in_sizes, int n_in,
                              void* d_out, int out_size, void* d_ws, size_t ws_size,
                              hipStream_t stream)
{
    const float* x      = (const float*)d_in[0];
    const int*   eidx   = (const int*)  d_in[1];
    const float* noise  = (const float*)d_in[2];
    const float* W_proj = (const float*)d_in[3];
    const float* b_proj = (const float*)d_in[4];
    const float* W_gat  = (const float*)d_in[5];
    const float* a_src  = (const float*)d_in[6];
    const float* a_dst  = (const float*)d_in[7];
    const float* b_gat  = (const float*)d_in[8];
    const float* g1     = (const float*)d_in[9];
    const float* be1    = (const float*)d_in[10];
    const float* W_qkv  = (const float*)d_in[11];
    const float* b_qkv  = (const float*)d_in[12];
    const float* W_o    = (const float*)d_in[13];
    const float* b_o    = (const float*)d_in[14];
    const float* g2     = (const float*)d_in[15];
    const float* be2    = (const float*)d_in[16];
    const float* Wc1    = (const float*)d_in[17];
    const float* bc1    = (const float*)d_in[18];
    const float* gc     = (const float*)d_in[19];
    const float* bec    = (const float*)d_in[20];
    const float* Wc2    = (const float*)d_in[21];
    const float* bc2    = (const float*)d_in[22];

    // workspace layout (floats); RT = N*T*HID = 5,120,000
    float* ws = (float*)d_ws;
    const size_t RT = (size_t)N_ * T_ * HID_;
    float*    regA   = ws;                       // h -> nrm -> ta   (RT)
    float*    feats  = ws + RT;                  // nf (N,T,HID)     (RT)
    float*    qkvbuf = ws + 2 * RT;              // qkv -> attn_proj (3*RT)
    float*    xh     = ws + 5 * RT;              // per-t (N,HID)
    float*    as     = xh + (size_t)N_ * HID_;   // (N,H)
    float*    ad     = as + (size_t)N_ * H_;
    unsigned* mkey   = (unsigned*)(ad + (size_t)N_ * H_);
    float*    ssum   = (float*)(mkey + (size_t)N_ * H_);
    float*    ebuf   = ssum + (size_t)N_ * H_;   // (ETOT,H)
    float*    h1     = ebuf + (size_t)ETOT_ * H_;
    float*    h1g    = h1 + (size_t)N_ * HID_;

    const int NT = N_ * T_;                      // 40000 rows
    const dim3 gb(128);

    // 1) h = relu(x @ W_proj + b_proj)     [40000 x 64 -> 128]
    gemm_f32_wmma<false, true><<<dim3((NT + 15) / 16, HID_ / 64), gb, 0, stream>>>(
        x, IN_DIM_, W_proj, HID_, b_proj, regA, HID_, NT, IN_DIM_);

    // 2) GAT per timestep -> feats (N,T,HID), pre-seeded with b_gat
    const int nft = N_ * T_ * HID_;
    feats_init_kernel<<<(nft + 255) / 256, 256, 0, stream>>>(feats, b_gat, nft);
    const int net = ETOT_ * H_;
    for (int t = 0; t < T_; ++t) {
        const float* h_t  = regA + (size_t)t * N_ * HID_;
        const int*   ei_t = eidx + (size_t)t * 2 * E_;
        gemm_f32_wmma<false, false><<<dim3((N_ + 15) / 16, HID_ / 64), gb, 0, stream>>>(
            h_t, HID_, W_gat, HID_, nullptr, xh, HID_, N_, HID_);
        attcoef_kernel<<<(N_ * H_ + 255) / 256, 256, 0, stream>>>(xh, a_src, a_dst, as, ad);
        zero_ms_kernel<<<(N_ * H_ + 255) / 256, 256, 0, stream>>>(mkey, ssum);
        edge_max_kernel<<<(net + 255) / 256, 256, 0, stream>>>(ei_t, as, ad, ebuf, mkey);
        edge_sum_kernel<<<(net + 255) / 256, 256, 0, stream>>>(ei_t, ebuf, mkey, ssum);
        edge_agg_kernel<<<(net + 255) / 256, 256, 0, stream>>>(ei_t, ebuf, ssum, xh, feats, t);
    }

    // 3) nrm = LN(feats; g1, be1)  -> regA
    ln_kernel<<<(NT + 7) / 8, 256, 0, stream>>>(feats, g1, be1, regA, NT);

    // 4) qkv = nrm @ W_qkv^T + b_qkv       [40000 x 128 -> 384]
    gemm_f32_wmma<true, false><<<dim3((NT + 15) / 16, 384 / 64), gb, 0, stream>>>(
        regA, HID_, W_qkv, HID_, b_qkv, qkvbuf, 384, NT, HID_);

    // 5) temporal attention -> ta in regA
    const int natt = N_ * H_ * T_;
    attention_kernel<<<(natt + 255) / 256, 256, 0, stream>>>(qkvbuf, regA);

    // 6) attn_proj = ta @ W_o^T + b_o -> qkvbuf (reuse)
    gemm_f32_wmma<true, false><<<dim3((NT + 15) / 16, HID_ / 64), gb, 0, stream>>>(
        regA, HID_, W_o, HID_, b_o, qkvbuf, HID_, NT, HID_);

    // 7) fused = LN(nf + attn_proj)[:, T-1, :] (+div +noise) -> emb (d_out[0:N*HID])
    float* emb = (float*)d_out;
    fuse_emb_kernel<<<(N_ + 7) / 8, 256, 0, stream>>>(feats, qkvbuf, g2, be2, noise, emb);

    // 8) h1 = emb @ Wc1 + bc1              [5000 x 128 -> 128]
    gemm_f32_wmma<false, false><<<dim3((N_ + 15) / 16, HID_ / 64), gb, 0, stream>>>(
        emb, HID_, Wc1, HID_, bc1, h1, HID_, N_, HID_);

    // 9) h1g = gelu(LN(h1; gc, bec))
    ln_gelu_kernel<<<(N_ + 7) / 8, 256, 0, stream>>>(h1, gc, bec, h1g, N_);

    // 10) logits = h1g @ Wc2 + bc2 (+sinu on col 1) -> d_out[N*HID:]
    logits_kernel<<<(N_ + 255) / 256, 256, 0, stream>>>(
        h1g, Wc2, bc2, emb + (size_t)N_ * HID_);
}